// GATClassifier_52621939310633
// MI455X (gfx1250) — compile-verified
//
#include <hip/hip_runtime.h>
#include <math.h>

// ---------------- problem constants (from reference) ----------------
#define N_NODES   50000
#define N_EDGES   1600000
#define IN_FEATS  128
#define HEADS     4
#define HID       32
#define C_OUT     (HEADS * HID)   // 128
#define N_GRAPHS  64
#define NEG_SLOPE 0.2f

typedef __attribute__((ext_vector_type(2))) float v2f;
typedef __attribute__((ext_vector_type(8))) float v8f;

// Monotone float<->int order-preserving encoding (for atomicMax on float values)
__device__ __forceinline__ int enc_f32(float f) {
  int i = __float_as_int(f);
  return i >= 0 ? i : (i ^ 0x7FFFFFFF);
}
__device__ __forceinline__ float dec_f32(int i) {
  int b = i >= 0 ? i : (i ^ 0x7FFFFFFF);
  return __int_as_float(b);
}
#define ENC_NEG_INF ((int)0x807FFFFF)   // enc(-inf)

__device__ __forceinline__ float leaky(float v) {
  return v > 0.0f ? v : NEG_SLOPE * v;
}

// ---------------- 0) init scratch ----------------
__global__ void gat_init(float* __restrict__ rst, float* __restrict__ denom,
                         int* __restrict__ m, float* __restrict__ hg,
                         float* __restrict__ counts) {
  int idx = blockIdx.x * blockDim.x + threadIdx.x;
  if (idx < N_NODES * C_OUT)  rst[idx]   = 0.0f;
  if (idx < N_NODES * HEADS) { denom[idx] = 0.0f; m[idx] = ENC_NEG_INF; }
  if (idx < N_GRAPHS * C_OUT) hg[idx]    = 0.0f;
  if (idx < N_GRAPHS)         counts[idx] = 0.0f;
}

// ---------------- 1) feat = features @ W via V_WMMA_F32_16X16X4_F32 ----------------
// One wave per 16x16 output tile. 8 waves/block cover one 16-row stripe across
// all 128 output columns. A: 16x4 f32 (2 VGPR/lane), B: 4x16 f32 (2 VGPR/lane),
// C/D: 16x16 f32 (8 VGPR/lane) per ISA layout.
__global__ __launch_bounds__(256)
void gat_gemm_wmma(const float* __restrict__ feats, const float* __restrict__ W,
                   float* __restrict__ feat) {
  const int lane = threadIdx.x & 31;
  const int wave = threadIdx.x >> 5;      // 0..7 -> column tile
  const int half = lane >> 4;             // 0/1
  const int l    = lane & 15;
  const int rowBase = blockIdx.x * 16;
  const int colBase = wave * 16;
  const int arow = rowBase + l;
  const int col  = colBase + l;

  v8f c = {};
  #pragma unroll
  for (int k = 0; k < IN_FEATS; k += 4) {
    // A fragment: lane l holds A[M=l][K=k+2*half .. +1]
    const float* ap = feats + (size_t)arow * IN_FEATS + k + half * 2;
    v2f a; a.x = ap[0]; a.y = ap[1];
    // B fragment: lane l holds B[K=k+2*half .. +1][N=col]  (W is [K=128][N=128] row-major)
    const int kr = k + half * 2;
    v2f b; b.x = W[kr * C_OUT + col]; b.y = W[(kr + 1) * C_OUT + col];
    c = __builtin_amdgcn_wmma_f32_16x16x4_f32(false, a, false, b, (short)0, c,
                                              false, false);
  }
  // D: lanes 0-15 -> M=0..7 in c[0..7], lanes 16-31 -> M=8..15
  #pragma unroll
  for (int i = 0; i < 8; ++i) {
    int row = rowBase + half * 8 + i;
    feat[(size_t)row * C_OUT + col] = c[i];
  }
}

// ---------------- 2) el/er per (node, head) ----------------
__global__ void gat_el_er(const float* __restrict__ feat,
                          const float* __restrict__ attn_l,
                          const float* __restrict__ attn_r,
                          float* __restrict__ el, float* __restrict__ er) {
  int idx = blockIdx.x * blockDim.x + threadIdx.x;
  if (idx >= N_NODES * HEADS) return;
  int n = idx >> 2, h = idx & 3;
  const float* f  = feat + (size_t)n * C_OUT + h * HID;
  const float* al = attn_l + h * HID;
  const float* ar = attn_r + h * HID;
  float sl = 0.0f, sr = 0.0f;
  #pragma unroll
  for (int d = 0; d < HID; ++d) { float v = f[d]; sl += v * al[d]; sr += v * ar[d]; }
  el[idx] = sl; er[idx] = sr;
}

// ---------------- 3) edge softmax: segment max ----------------
__global__ void gat_edge_max(const int* __restrict__ src, const int* __restrict__ dst,
                             const float* __restrict__ el, const float* __restrict__ er,
                             int* __restrict__ m) {
  int idx = blockIdx.x * blockDim.x + threadIdx.x;
  if (idx >= N_EDGES * HEADS) return;
  int e = idx >> 2, h = idx & 3;
  int s = src[e], d = dst[e];
  float v = leaky(el[s * HEADS + h] + er[d * HEADS + h]);
  atomicMax(&m[d * HEADS + h], enc_f32(v));
}

// ---------------- 4) edge softmax: segment sum of exp ----------------
__global__ void gat_edge_sum(const int* __restrict__ src, const int* __restrict__ dst,
                             const float* __restrict__ el, const float* __restrict__ er,
                             const int* __restrict__ m, float* __restrict__ denom) {
  int idx = blockIdx.x * blockDim.x + threadIdx.x;
  if (idx >= N_EDGES * HEADS) return;
  int e = idx >> 2, h = idx & 3;
  int s = src[e], d = dst[e];
  float v  = leaky(el[s * HEADS + h] + er[d * HEADS + h]);
  float mm = dec_f32(m[d * HEADS + h]);
  atomicAdd(&denom[d * HEADS + h], expf(v - mm));
}

// ---------------- 5) weighted aggregation rst[dst] += a * feat[src] ----------------
__global__ void gat_edge_aggr(const int* __restrict__ src, const int* __restrict__ dst,
                              const float* __restrict__ el, const float* __restrict__ er,
                              const int* __restrict__ m, const float* __restrict__ denom,
                              const float* __restrict__ feat, float* __restrict__ rst) {
  int idx = blockIdx.x * blockDim.x + threadIdx.x;
  if (idx >= N_EDGES * HEADS) return;
  int e = idx >> 2, h = idx & 3;
  int s = src[e], d = dst[e];
  float v  = leaky(el[s * HEADS + h] + er[d * HEADS + h]);
  float mm = dec_f32(m[d * HEADS + h]);
  float a  = expf(v - mm) / denom[d * HEADS + h];
  const float* fs = feat + (size_t)s * C_OUT + h * HID;
  float*       rd = rst  + (size_t)d * C_OUT + h * HID;
  #pragma unroll 8
  for (int d0 = 0; d0 < HID; ++d0) atomicAdd(&rd[d0], fs[d0] * a);
}

// ---------------- 6) per-graph node counts ----------------
__global__ void gat_counts(const int* __restrict__ gid, float* __restrict__ counts) {
  int n = blockIdx.x * blockDim.x + threadIdx.x;
  if (n >= N_NODES) return;
  atomicAdd(&counts[gid[n]], 1.0f);
}

// ---------------- 7) ELU(rst + bias) + graph-sum pooling ----------------
__global__ void gat_pool(const float* __restrict__ rst, const float* __restrict__ bias,
                         const int* __restrict__ gid, float* __restrict__ hg) {
  int idx = blockIdx.x * blockDim.x + threadIdx.x;
  if (idx >= N_NODES * C_OUT) return;
  int n = idx >> 7, c = idx & 127;
  float v = rst[idx] + bias[c];
  v = v > 0.0f ? v : expm1f(v);          // ELU(alpha=1)
  atomicAdd(&hg[gid[n] * C_OUT + c], v);
}

// ---------------- 8) mean + linear + sigmoid ----------------
__global__ void gat_final(const float* __restrict__ hg, const float* __restrict__ counts,
                          const float* __restrict__ fc_w, const float* __restrict__ fc_b,
                          float* __restrict__ out) {
  int g = blockIdx.x * blockDim.x + threadIdx.x;
  if (g >= N_GRAPHS) return;
  float cnt = counts[g];
  float inv = cnt > 0.0f ? 1.0f / cnt : 0.0f;
  float acc = fc_b[0];
  for (int c = 0; c < C_OUT; ++c) acc += hg[g * C_OUT + c] * inv * fc_w[c];
  out[g] = 1.0f / (1.0f + expf(-acc));
}

// ---------------- workspace layout (bytes) ----------------
#define OFF_FEAT   ((size_t)0)                                   // N*128 f32 = 25.6 MB
#define OFF_EL     (OFF_FEAT  + (size_t)N_NODES * C_OUT * 4)     // N*4 f32
#define OFF_ER     (OFF_EL    + (size_t)N_NODES * HEADS * 4)
#define OFF_M      (OFF_ER    + (size_t)N_NODES * HEADS * 4)     // int-encoded max
#define OFF_DENOM  (OFF_M     + (size_t)N_NODES * HEADS * 4)
#define OFF_RST    (OFF_DENOM + (size_t)N_NODES * HEADS * 4)     // N*128 f32 = 25.6 MB
#define OFF_HG     (OFF_RST   + (size_t)N_NODES * C_OUT * 4)     // 64*128 f32
#define OFF_CNT    (OFF_HG    + (size_t)N_GRAPHS * C_OUT * 4)    // 64 f32

extern "C" void kernel_launch(void* const* d_in, const int* in_sizes, int n_in,
                              void* d_out, int out_size, void* d_ws, size_t ws_size,
                              hipStream_t stream) {
  const float* features = (const float*)d_in[0];
  const int*   src      = (const int*)  d_in[1];
  const int*   dst      = (const int*)  d_in[2];
  const int*   gid      = (const int*)  d_in[3];
  const float* W        = (const float*)d_in[4];
  const float* attn_l   = (const float*)d_in[5];
  const float* attn_r   = (const float*)d_in[6];
  const float* bias     = (const float*)d_in[7];
  const float* fc_w     = (const float*)d_in[8];
  const float* fc_b     = (const float*)d_in[9];
  float* out = (float*)d_out;

  char* ws = (char*)d_ws;
  float* feat   = (float*)(ws + OFF_FEAT);
  float* el     = (float*)(ws + OFF_EL);
  float* er     = (float*)(ws + OFF_ER);
  int*   m      = (int*)  (ws + OFF_M);
  float* denom  = (float*)(ws + OFF_DENOM);
  float* rst    = (float*)(ws + OFF_RST);
  float* hg     = (float*)(ws + OFF_HG);
  float* counts = (float*)(ws + OFF_CNT);

  const int threads = 256;
  const int nc_blocks = (N_NODES * C_OUT + threads - 1) / threads;   // 25000
  const int eh_blocks = (N_EDGES * HEADS + threads - 1) / threads;   // 25000
  const int nh_blocks = (N_NODES * HEADS + threads - 1) / threads;

  // 0) init accumulators / -inf maxes
  gat_init<<<nc_blocks, threads, 0, stream>>>(rst, denom, m, hg, counts);

  // 1) GEMM via fp32 WMMA: 3125 row-stripes x (8 waves = 8 col tiles)
  gat_gemm_wmma<<<N_NODES / 16, 256, 0, stream>>>(features, W, feat);

  // 2) attention logits per node/head
  gat_el_er<<<nh_blocks, threads, 0, stream>>>(feat, attn_l, attn_r, el, er);

  // 3-5) edge softmax + aggregation
  gat_edge_max <<<eh_blocks, threads, 0, stream>>>(src, dst, el, er, m);
  gat_edge_sum <<<eh_blocks, threads, 0, stream>>>(src, dst, el, er, m, denom);
  gat_edge_aggr<<<eh_blocks, threads, 0, stream>>>(src, dst, el, er, m, denom, feat, rst);

  // 6-7) readout
  gat_counts<<<(N_NODES + threads - 1) / threads, threads, 0, stream>>>(gid, counts);
  gat_pool  <<<nc_blocks, threads, 0, stream>>>(rst, bias, gid, hg);

  // 8) classifier
  gat_final<<<1, 64, 0, stream>>>(hg, counts, fc_w, fc_b, out);
}